// TA_GRU_82059645157888
// MI455X (gfx1250) — compile-verified
//
#include <hip/hip_runtime.h>
#include <math.h>

// ---- model dims ----
#define WE   300
#define TE   10
#define SL   24
#define NT1  6
#define UT   3
#define HL   75          // H_LSTM
#define HG   77          // H_GRU
#define CLS  27
#define DIN  7210        // SL*WE + TE
#define NSEQ 4608        // 256*3*6
#define BT   768         // 256*3
#define NB   256
#define ZN   304         // 4*HL padded to 19*16
#define KPAD 76          // HL padded to multiple of 4
#define XWT  384         // floats per (n,ntile) swizzled tile: 256 (mtile0) + 128 (mtile1)

// ---- workspace layout (float offsets); needs ~138 MB ----
static const size_t OFF_XW  = 0;                               // NSEQ*19*XWT = 33,619,968
static const size_t OFF_HM  = (size_t)NSEQ * 19 * XWT;
static const size_t OFF_CTX = OFF_HM  + (size_t)NSEQ * HL;
static const size_t OFF_XG  = OFF_CTX + (size_t)BT * (WE + TE);
static const size_t OFF_SEQ = OFF_XG  + (size_t)BT * (3 * HG);

typedef float v2f __attribute__((ext_vector_type(2)));
typedef float v8f __attribute__((ext_vector_type(8)));

__device__ __forceinline__ float sigf(float x) { return 1.0f / (1.0f + expf(-x)); }

// ============ Kernel 1: XW[n] = we[n] @ lstm_W + lstm_b  (f32 WMMA) ============
// grid NSEQ, block 256 (8 waves). Output stored in WMMA C-register swizzled
// layout: per (n,ntile): [0..255] = mtile0 lane-major 8 each; [256..383] =
// mtile1 (rows 16..23) l16-major 8 each.
__global__ __launch_bounds__(256) void k1_xw(const float* __restrict__ inp,
                                             const float* __restrict__ W,
                                             const float* __restrict__ bias,
                                             float* __restrict__ xw) {
  __shared__ float sA[SL * WE];
  const int n = blockIdx.x, tid = threadIdx.x;
  const float* src = inp + (size_t)n * DIN;
  for (int i = tid; i < SL * WE; i += 256) sA[i] = src[i];
  __syncthreads();

  const int lane = tid & 31, wave = tid >> 5;
  const int l16 = lane & 15;
  const bool hi = lane >= 16;
  const int mb  = (wave & 1) * 16;            // a wave's tiles share one mtile
  const int nt0 = wave >> 1;                  // ntiles nt0,+4,+8,+12,+16 (clamped)

  int ntile[5], col[5], colc[5];
  bool tvalid[5];
  for (int t = 0; t < 5; ++t) {
    const int raw = nt0 + 4 * t;
    tvalid[t] = raw < 19;                     // wave-uniform
    ntile[t]  = tvalid[t] ? raw : 18;         // clamp (redundant compute, store skipped)
    col[t]    = ntile[t] * 16 + l16;          // <= 303
    colc[t]   = col[t] < WE ? col[t] : WE - 1;
  }

  v8f acc[5];
  for (int t = 0; t < 5; ++t)
    for (int v = 0; v < 8; ++v) acc[t][v] = 0.0f;

  const int arow  = mb + l16;
  const bool av   = arow < SL;
  const int arowc = av ? arow : SL - 1;

  for (int kc = 0; kc < WE; kc += 4) {
    const int k0 = kc + (hi ? 2 : 0);
    const v2f al = *(const v2f*)&sA[arowc * WE + k0];   // ds_load_b64
    v2f a;
    a.x = av ? al.x : 0.0f;
    a.y = av ? al.y : 0.0f;
    for (int t = 0; t < 5; ++t) {
      const bool cv = col[t] < WE;
      const float b0 = W[(size_t)k0 * WE + colc[t]];
      const float b1 = W[(size_t)(k0 + 1) * WE + colc[t]];
      v2f b;
      b.x = cv ? b0 : 0.0f;
      b.y = cv ? b1 : 0.0f;
      acc[t] = __builtin_amdgcn_wmma_f32_16x16x4_f32(false, a, false, b,
                                                     (short)0, acc[t], false, false);
    }
  }
  for (int t = 0; t < 5; ++t) {
    if (!tvalid[t]) continue;                 // wave-uniform branch
    const float bs = (col[t] < WE) ? bias[colc[t]] : 0.0f;
    v8f val;
    for (int v = 0; v < 8; ++v) val[v] = acc[t][v] + bs;
    float* dst = xw + ((size_t)n * 19 + ntile[t]) * XWT;
    if (mb == 0) {
      *(v8f*)&dst[lane * 8] = val;            // 2x global_store_b128
    } else if (!hi) {
      *(v8f*)&dst[256 + l16 * 8] = val;       // rows 16..23 only
    }
  }
}

// ============ Kernel 2: sequential LSTM over 4608 steps (1 WGP, WMMA) ============
// grid 1, block 512 (16 waves). Fixed 3 tiles/wave (clamped). U pair-interleaved
// in LDS for b64 B fetch; XW read in swizzled C layout (unconditional b128).
__global__ __launch_bounds__(512) void k2_lstm(const float* __restrict__ xw,
                                               const float* __restrict__ U,
                                               float* __restrict__ hmean) {
  __shared__ float sUp[KPAD * ZN];  // pair-interleaved: [(k>>1)*2*ZN + c*2 + (k&1)]
  __shared__ float sH[32 * KPAD];
  __shared__ float sC[32 * KPAD];
  __shared__ float sZ[32 * ZN];
  const int tid = threadIdx.x;
  for (int i = tid; i < KPAD * ZN; i += 512) {
    const int k = i / ZN, c = i % ZN;
    const float v = (k < HL && c < WE) ? U[(size_t)k * WE + c] : 0.0f;
    sUp[(k >> 1) * (2 * ZN) + c * 2 + (k & 1)] = v;
  }
  for (int i = tid; i < 32 * KPAD; i += 512) { sH[i] = 0.0f; sC[i] = 0.0f; }
  __syncthreads();

  const int lane = tid & 31, wave = tid >> 5;
  const int l16 = lane & 15;
  const bool hi = lane >= 16;
  const int mb  = (wave & 1) * 16;
  const int nt0 = wave >> 1;                  // ntiles nt0,+8,+16 (clamped)

  int ntc[3], col[3];
  bool tvalid[3];
  for (int t = 0; t < 3; ++t) {
    const int raw = nt0 + 8 * t;
    tvalid[t] = raw < 19;                     // wave-uniform
    ntc[t] = tvalid[t] ? raw : 18;
    col[t] = ntc[t] * 16 + l16;               // <= 303
  }
  const int arow = mb + l16;
  // swizzled-XW lane offset within a tile (mtile1 hi lanes read duplicate data;
  // it only feeds pad rows 24..31 of D, which are never consumed)
  const int xwoff = (mb == 0) ? (lane * 8) : (256 + l16 * 8);

  for (int n = 0; n < NSEQ; ++n) {
    const float* xb = xw + (size_t)n * 19 * XWT;
    v8f acc[3];
    for (int t = 0; t < 3; ++t)
      acc[t] = *(const v8f*)&xb[ntc[t] * XWT + xwoff];   // 2x global_load_b128
    for (int kc = 0; kc < KPAD; kc += 4) {    // 19 chunks, K padded to 76
      const int k0 = kc + (hi ? 2 : 0);
      const int p  = (k0 >> 1) * (2 * ZN);
      const v2f a = *(const v2f*)&sH[arow * KPAD + k0];        // ds_load_b64
      for (int t = 0; t < 3; ++t) {
        const v2f b = *(const v2f*)&sUp[p + col[t] * 2];       // ds_load_b64
        acc[t] = __builtin_amdgcn_wmma_f32_16x16x4_f32(false, a, false, b,
                                                       (short)0, acc[t], false, false);
      }
    }
    for (int t = 0; t < 3; ++t) {
      if (!tvalid[t]) continue;               // wave-uniform
      for (int v = 0; v < 8; ++v) {
        const int row = mb + v + (hi ? 8 : 0);
        sZ[row * ZN + col[t]] = acc[t][v];    // pad rows (24..31) harmless
      }
    }
    __syncthreads();
    // gates + state update: 24*75 cells; gate order i,f,g,o
    for (int j = tid; j < SL * HL; j += 512) {
      const int s = j / HL, q = j % HL;
      const float zi = sZ[s * ZN + q];
      const float zf = sZ[s * ZN + HL + q];
      const float zg = sZ[s * ZN + 2 * HL + q];
      const float zo = sZ[s * ZN + 3 * HL + q];
      const float cn = sigf(zf) * sC[s * KPAD + q] + sigf(zi) * tanhf(zg);
      sC[s * KPAD + q] = cn;
      sH[s * KPAD + q] = sigf(zo) * tanhf(cn);
    }
    __syncthreads();
    for (int q = tid; q < HL; q += 512) {
      float m = 0.0f;
      for (int s = 0; s < SL; ++s) m += sH[s * KPAD + q];
      hmean[(size_t)n * HL + q] = m * (1.0f / 24.0f);
    }
  }
}

// ============ Kernel 3: ctx rows (folded tw/comb/einsum algebra) ============
__global__ __launch_bounds__(320) void k3_ctx(const float* __restrict__ hm,
                                              const float* __restrict__ twW,
                                              const float* __restrict__ twb,
                                              const float* __restrict__ A,
                                              const float* __restrict__ Bt,
                                              const float* __restrict__ inp,
                                              float* __restrict__ ctx) {
  __shared__ float g[HL];
  const int bt = blockIdx.x, tid = threadIdx.x;
  if (tid < HL) {
    float a = 0.0f;
    for (int nn = 0; nn < NT1; ++nn)
      a += A[nn] * hm[((size_t)bt * NT1 + nn) * HL + tid];
    g[tid] = a;
  }
  __syncthreads();
  if (tid < WE + TE) {
    const float bb = Bt[0];
    float out;
    if (tid < WE) {
      float a = 0.0f;
      for (int k = 0; k < HL; ++k) a += g[k] * twW[(size_t)k * WE + tid];
      float sA = 0.0f;
      for (int nn = 0; nn < NT1; ++nn) sA += A[nn];
      out = (1000.0f / 1001.0f) * (a + sA * twb[tid]) + bb;
    } else {
      const int j = tid - WE;
      float a = 0.0f;
      for (int nn = 0; nn < NT1; ++nn)
        a += A[nn] * inp[((size_t)bt * NT1 + nn) * DIN + SL * WE + j];
      out = a * (1.0f / 1001.0f) + bb;
    }
    ctx[(size_t)bt * (WE + TE) + tid] = out;
  }
}

// ============ Kernel 4: xg = ctx @ gru_W + gru_b[0] ============
__global__ __launch_bounds__(256) void k4_xg(const float* __restrict__ ctx,
                                             const float* __restrict__ gW,
                                             const float* __restrict__ gb,
                                             float* __restrict__ xg) {
  __shared__ float sc[WE + TE];
  const int bt = blockIdx.x, tid = threadIdx.x;
  for (int i = tid; i < WE + TE; i += 256) sc[i] = ctx[(size_t)bt * (WE + TE) + i];
  __syncthreads();
  if (tid < 3 * HG) {
    float a = gb[tid];
    for (int k = 0; k < WE + TE; ++k) a += sc[k] * gW[(size_t)k * (3 * HG) + tid];
    xg[(size_t)bt * (3 * HG) + tid] = a;
  }
}

// ============ Kernel 5: sequential GRU over 256 steps, "batch"=UT=3 ============
__global__ __launch_bounds__(256) void k5_gru(const float* __restrict__ xg,
                                              const float* __restrict__ Ug,
                                              const float* __restrict__ gb,
                                              float* __restrict__ seq) {
  __shared__ float h[UT * HG];
  __shared__ float rg[UT * 3 * HG];
  const int tid = threadIdx.x;
  for (int i = tid; i < UT * HG; i += 256) h[i] = 0.0f;
  __syncthreads();
  for (int s = 0; s < NB; ++s) {
    for (int j = tid; j < UT * 3 * HG; j += 256) {
      const int t = j / (3 * HG), f = j % (3 * HG);
      float a = gb[3 * HG + f];
      for (int k = 0; k < HG; ++k) a += h[t * HG + k] * Ug[(size_t)k * (3 * HG) + f];
      rg[j] = a;
    }
    __syncthreads();
    if (tid < UT * HG) {
      const int t = tid / HG, k = tid % HG;
      const float* xr = xg + ((size_t)s * UT + t) * (3 * HG);
      const float z  = sigf(xr[k]          + rg[t * (3 * HG) + k]);
      const float r  = sigf(xr[HG + k]     + rg[t * (3 * HG) + HG + k]);
      const float hh = tanhf(xr[2 * HG + k] + r * rg[t * (3 * HG) + 2 * HG + k]);
      const float hn = z * h[tid] + (1.0f - z) * hh;
      h[tid] = hn;
      if (t == UT - 1) seq[(size_t)s * HG + k] = hn;
    }
    __syncthreads();
  }
}

// ============ Kernel 6: logits + softmax ============
__global__ __launch_bounds__(32) void k6_out(const float* __restrict__ seq,
                                             const float* __restrict__ lW,
                                             const float* __restrict__ lb,
                                             float* __restrict__ out) {
  __shared__ float sl[CLS];
  const int s = blockIdx.x, tid = threadIdx.x;
  if (tid < CLS) {
    float a = lb[tid];
    for (int k = 0; k < HG; ++k) a += seq[(size_t)s * HG + k] * lW[k * CLS + tid];
    sl[tid] = a;
  }
  __syncthreads();
  if (tid < CLS) {
    float m = sl[0];
    for (int k = 1; k < CLS; ++k) m = fmaxf(m, sl[k]);
    float ssum = 0.0f;
    for (int k = 0; k < CLS; ++k) ssum += expf(sl[k] - m);
    out[(size_t)s * CLS + tid] = expf(sl[tid] - m) / ssum;
  }
}

extern "C" void kernel_launch(void* const* d_in, const int* in_sizes, int n_in,
                              void* d_out, int out_size, void* d_ws, size_t ws_size,
                              hipStream_t stream) {
  const float* inp   = (const float*)d_in[0];
  const float* lstmW = (const float*)d_in[1];
  const float* lstmU = (const float*)d_in[2];
  const float* lstmb = (const float*)d_in[3];
  const float* twW   = (const float*)d_in[4];
  const float* twb   = (const float*)d_in[5];
  const float* Atw   = (const float*)d_in[6];
  const float* Btw   = (const float*)d_in[7];
  const float* gW    = (const float*)d_in[8];
  const float* gU    = (const float*)d_in[9];
  const float* gb    = (const float*)d_in[10];
  const float* lW    = (const float*)d_in[11];
  const float* lb    = (const float*)d_in[12];

  float* ws  = (float*)d_ws;
  float* xw  = ws + OFF_XW;
  float* hm  = ws + OFF_HM;
  float* ctx = ws + OFF_CTX;
  float* xgp = ws + OFF_XG;
  float* seq = ws + OFF_SEQ;

  k1_xw  <<<NSEQ, 256, 0, stream>>>(inp, lstmW, lstmb, xw);
  k2_lstm<<<1,    512, 0, stream>>>(xw, lstmU, hm);
  k3_ctx <<<BT,   320, 0, stream>>>(hm, twW, twb, Atw, Btw, inp, ctx);
  k4_xg  <<<BT,   256, 0, stream>>>(ctx, gW, gb, xgp);
  k5_gru <<<1,    256, 0, stream>>>(xgp, gU, gb, seq);
  k6_out <<<NB,    32, 0, stream>>>(seq, lW, lb, (float*)d_out);
}